// verletNetworks_28527172780521
// MI455X (gfx1250) — compile-verified
//
#include <hip/hip_runtime.h>
#include <hip/hip_bf16.h>
#include <math.h>

typedef __attribute__((ext_vector_type(2))) float v2f;
typedef __attribute__((ext_vector_type(8))) float v8f;

#define H_STEP 0.1f
#define TV_EPS 1e-3f

// ---------------------------------------------------------------------------
// Store one 64(M) x 16(N) strip of accumulators.
// MODE 0: Y[64,N] = acc            (channel-major)
// MODE 1: Y[N,64] = acc^T          (node-major, for fast per-node gather)
// MODE 2: Y[64,N] += alpha * acc   (Verlet node update)
// ---------------------------------------------------------------------------
template<int MODE>
__device__ __forceinline__ void vn_store_strip(float* Y, long col, long N, int hf,
                                               const v8f* acc, float alpha)
{
    if (MODE == 1) {
        float* yb = Y + (size_t)col * 64 + 8 * hf;   // two contiguous 32B runs/tile
#pragma unroll
        for (int t = 0; t < 4; ++t)
#pragma unroll
            for (int r = 0; r < 8; ++r)
                yb[t * 16 + r] = acc[t][r];
    } else if (MODE == 2) {
#pragma unroll
        for (int t = 0; t < 4; ++t)
#pragma unroll
            for (int r = 0; r < 8; ++r) {
                size_t idx = (size_t)(t * 16 + 8 * hf + r) * N + col;
                Y[idx] += alpha * acc[t][r];
            }
    } else {
#pragma unroll
        for (int t = 0; t < 4; ++t)
#pragma unroll
            for (int r = 0; r < 8; ++r)
                Y[(size_t)(t * 16 + 8 * hf + r) * N + col] = acc[t][r];
    }
}

// ---------------------------------------------------------------------------
// Y = W(64 x K) @ X(K x N) via V_WMMA_F32_16X16X4_F32.
// Block = 128 threads = 4 waves; each wave owns a 64(M) x 32(N) tile
// (two 16-col strips sharing the A fragments -> 8 WMMAs per A load).
// K is compile-time: full unroll, clause-batched loads.
// ---------------------------------------------------------------------------
template<int K, int MODE>
__global__ __launch_bounds__(128)
void vn_gemm(const float* __restrict__ W, const float* X, float* Y,
             long N, float alpha)
{
    const int lane = threadIdx.x & 31;
    const int wv   = threadIdx.x >> 5;
    const int l2   = lane & 15;       // row/col within half
    const int hf   = lane >> 4;       // lane half -> K split {K0,K1 | K2,K3}
    const long n0  = ((long)blockIdx.x * 4 + wv) * 32;
    if (n0 >= N) return;              // wave-uniform: EXEC stays all-ones
    const bool two = (n0 + 32 <= N);  // second strip present (N % 16 == 0)
    const long c0  = n0 + l2;
    const long c1  = c0 + 16;

    v8f acc0[4] = {v8f{0.f}, v8f{0.f}, v8f{0.f}, v8f{0.f}};
    v8f acc1[4] = {v8f{0.f}, v8f{0.f}, v8f{0.f}, v8f{0.f}};

#pragma unroll
    for (int k0 = 0; k0 < K; k0 += 4) {
        const int kk = k0 + 2 * hf;
        v2f a[4];
#pragma unroll
        for (int t = 0; t < 4; ++t) {      // 4 stacked M tiles: rows 16t..16t+15
            const float* wp = W + (size_t)(t * 16 + l2) * K + kk;
            a[t][0] = wp[0];
            a[t][1] = wp[1];
        }
        v2f b0;
        b0[0] = X[(size_t)kk * N + c0];
        b0[1] = X[(size_t)(kk + 1) * N + c0];
#pragma unroll
        for (int t = 0; t < 4; ++t)
            acc0[t] = __builtin_amdgcn_wmma_f32_16x16x4_f32(
                false, a[t], false, b0, (short)0, acc0[t], false, false);
        if (two) {
            v2f b1;
            b1[0] = X[(size_t)kk * N + c1];
            b1[1] = X[(size_t)(kk + 1) * N + c1];
#pragma unroll
            for (int t = 0; t < 4; ++t)
                acc1[t] = __builtin_amdgcn_wmma_f32_16x16x4_f32(
                    false, a[t], false, b1, (short)0, acc1[t], false, false);
        }
    }

    vn_store_strip<MODE>(Y, c0, N, hf, acc0, alpha);
    if (two)
        vn_store_strip<MODE>(Y, c1, N, hf, acc1, alpha);
}

// ---------------------------------------------------------------------------
// Per-edge: gather node features (node-major), grad/ave, tv_norm over 64
// channels, relu, xe += H * Ai.   One thread per edge.
// ---------------------------------------------------------------------------
__global__ __launch_bounds__(256)
void vn_edge_update(const float* __restrict__ G /*[NN,64]*/,
                    const int* __restrict__ iI, const int* __restrict__ jI,
                    float* xe /*[64,NE]*/, int NE)
{
    int e = blockIdx.x * blockDim.x + threadIdx.x;
    if (e >= NE) return;
    const float* gi = G + (size_t)iI[e] * 64;
    const float* gj = G + (size_t)jI[e] * 64;

    float v[64];
    float s = 0.f;
#pragma unroll
    for (int c = 0; c < 32; ++c) {          // grad channels (KN rows)
        float t = gi[c] - gj[c];
        v[c] = t; s += t;
    }
#pragma unroll
    for (int c = 32; c < 64; ++c) {         // ave channels (KNa rows)
        float t = 0.5f * (gi[c] + gj[c]);
        v[c] = t; s += t;
    }
    float mean = s * (1.f / 64.f);
    float ss = 0.f;
#pragma unroll
    for (int c = 0; c < 64; ++c) {
        float t = v[c] - mean;
        ss += t * t;
    }
    float inv = 1.f / sqrtf(ss + TV_EPS);
#pragma unroll
    for (int c = 0; c < 64; ++c) {
        float t = (v[c] - mean) * inv;
        t = t > 0.f ? t : 0.f;
        size_t idx = (size_t)c * NE + e;
        xe[idx] += H_STEP * t;
    }
}

// ---------------------------------------------------------------------------
// edge_div / edge_ave scatter into DA = [div(64,NN) ; ave(64,NN)] via f32
// atomics (DA is 25.6 MB -> L2-resident).
// ---------------------------------------------------------------------------
__global__ __launch_bounds__(256)
void vn_edge_scatter(const float* __restrict__ xe, const int* __restrict__ iI,
                     const int* __restrict__ jI, float* DA, int NE, int NN)
{
    int e = blockIdx.x * blockDim.x + threadIdx.x;
    if (e >= NE) return;
    const int i = iI[e], j = jI[e];
    float* divb = DA;
    float* aveb = DA + (size_t)64 * NN;
#pragma unroll 8
    for (int c = 0; c < 64; ++c) {
        float g = xe[(size_t)c * NE + e];
        atomicAdd(&divb[(size_t)c * NN + i],  g);
        atomicAdd(&divb[(size_t)c * NN + j], -g);
        float h = 0.5f * g;
        atomicAdd(&aveb[(size_t)c * NN + i], h);
        atomicAdd(&aveb[(size_t)c * NN + j], h);
    }
}

__global__ __launch_bounds__(256)
void vn_zero(float* p, long n)
{
    long t = (long)blockIdx.x * blockDim.x + threadIdx.x;
    if (t < n) p[t] = 0.f;
}

// Pack Wn = [KN_l ; KNa_l] (64x64) and We = [KE_l | KEa_l] (64x128)
__global__ __launch_bounds__(256)
void vn_pack(const float* __restrict__ KNl, const float* __restrict__ KNal,
             const float* __restrict__ KEl, const float* __restrict__ KEal,
             float* Wn, float* We)
{
    int t = blockIdx.x * blockDim.x + threadIdx.x;
    if (t < 4096) {
        int m = t >> 6, k = t & 63;
        Wn[t] = (m < 32) ? KNl[m * 64 + k] : KNal[(m - 32) * 64 + k];
    } else if (t < 4096 + 8192) {
        int u = t - 4096;
        int m = u >> 7, k = u & 127;
        We[u] = (k < 64) ? KEl[m * 64 + k] : KEal[m * 64 + (k - 64)];
    }
}

extern "C" void kernel_launch(void* const* d_in, const int* in_sizes, int n_in,
                              void* d_out, int out_size, void* d_ws, size_t ws_size,
                              hipStream_t stream)
{
    const float* xn_in   = (const float*)d_in[0];
    const float* xe_in   = (const float*)d_in[1];
    const int*   iInd    = (const int*)d_in[2];
    const int*   jInd    = (const int*)d_in[3];
    const float* KNopen  = (const float*)d_in[4];
    const float* KEopen  = (const float*)d_in[5];
    const float* KNclose = (const float*)d_in[6];
    const float* KEclose = (const float*)d_in[7];
    const float* KN      = (const float*)d_in[8];   // [3,32,64]
    const float* KE      = (const float*)d_in[9];   // [3,64,64]
    const float* KNa     = (const float*)d_in[10];  // [3,32,64]
    const float* KEa     = (const float*)d_in[11];  // [3,64,64]

    const long NN = in_sizes[0] / 64;
    const long NE = in_sizes[2];
    const int nlayer = in_sizes[8] / (32 * 64);

    float* XN = (float*)d_out;                 // running xn [64,NN], also final
    float* XE = XN + (size_t)64 * NN;          // running xe [64,NE], also final

    float* ws  = (float*)d_ws;
    float* gga = ws;                           // [NN,64] node-major features
    float* DA  = gga + (size_t)NN * 64;        // [128,NN] = [div;ave]
    float* Wn  = DA + (size_t)128 * NN;        // 64x64
    float* We  = Wn + 4096;                    // 64x128

    const int gN  = (int)((NN + 127) / 128);   // GEMM blocks (128 cols/block)
    const int gE  = (int)((NE + 127) / 128);
    const int eB  = (int)((NE + 255) / 256);   // per-edge kernels
    const long daN = (long)128 * NN;
    const int zB  = (int)((daN + 255) / 256);

    // open: xn = KNopen @ xn_in ; xe = KEopen @ xe_in
    vn_gemm<64, 0><<<gN, 128, 0, stream>>>(KNopen, xn_in, XN, NN, 1.f);
    vn_gemm<64, 0><<<gE, 128, 0, stream>>>(KEopen, xe_in, XE, NE, 1.f);

    for (int l = 0; l < nlayer; ++l) {
        vn_pack<<<48, 256, 0, stream>>>(KN + (size_t)l * 2048, KNa + (size_t)l * 2048,
                                        KE + (size_t)l * 4096, KEa + (size_t)l * 4096,
                                        Wn, We);
        vn_zero<<<zB, 256, 0, stream>>>(DA, daN);
        // gga[NN,64] = ([KN;KNa] @ xn)^T   (transposed store for gather)
        vn_gemm<64, 1><<<gN, 128, 0, stream>>>(Wn, XN, gga, NN, 1.f);
        // xe += H * relu(tv_norm([grad;ave]))
        vn_edge_update<<<eB, 256, 0, stream>>>(gga, iInd, jInd, XE, (int)NE);
        // DA = [edge_div(xe) ; edge_ave(xe)]
        vn_edge_scatter<<<eB, 256, 0, stream>>>(XE, iInd, jInd, DA, (int)NE, (int)NN);
        // xn += H * ([KE|KEa] @ DA)
        vn_gemm<128, 2><<<gN, 128, 0, stream>>>(We, DA, XN, NN, H_STEP);
    }

    // close (in-place safe: each wave reads its 32 columns fully before storing)
    vn_gemm<64, 0><<<gN, 128, 0, stream>>>(KNclose, XN, XN, NN, 1.f);
    vn_gemm<64, 0><<<gE, 128, 0, stream>>>(KEclose, XE, XE, NE, 1.f);
}